// LinearAttention_21809843929139
// MI455X (gfx1250) — compile-verified
//
#include <hip/hip_runtime.h>
#include <cstdint>

// ---------------------------------------------------------------------------
// Linformer attention on MI455X (gfx1250), full-FP32 via V_WMMA_F32_16X16X4_F32.
// Memory-bound (~0.5GB traffic vs ~120 GFLOP at 23.3 TB/s), so fp32 matrix ops
// keep reference precision for free. GEMM tiles are staged by the Tensor Data
// Mover (tensor_load_to_lds + s_wait_tensorcnt), double-buffered so the DMA
// overlaps the WMMA stream.
// ---------------------------------------------------------------------------

typedef __attribute__((ext_vector_type(2))) float v2f;
typedef __attribute__((ext_vector_type(8))) float v8f;
typedef __attribute__((ext_vector_type(4))) unsigned int u32x4;
typedef __attribute__((ext_vector_type(8))) int i32x8;
typedef __attribute__((ext_vector_type(4))) int i32x4;

// D = A(16x4) * B(4x16) + C(16x16), fp32, wave32.
#define WMMA_F32(a, b, c)                                                     \
  __builtin_amdgcn_wmma_f32_16x16x4_f32(false, (a), false, (b), (short)0,     \
                                        (c), false, false)

constexpr int BATCH = 4, SEQ = 4096, DMODEL = 1024, NH = 16, DHEAD = 64, KDIM = 128;
constexpr size_t MN = (size_t)BATCH * SEQ * DMODEL;   // floats per Q/K/V

// ---------------------------------------------------------------------------
// TDM: issue one 2D tile load (global -> LDS). Descriptor per ISA 8.3/8.4:
//  group0: count=1 | lds_addr | global_addr(57b) | type=2
//  group1: data_size=4B, optional LDS pad (+1 DWORD per 32 DWORDs -> row
//          stride 33 floats, conflict-free for the 16-lane fragment reads),
//          tensor dims (kept generously in-bounds), tile WxH, row stride.
// ---------------------------------------------------------------------------
__device__ inline void tdm_load_2d(uint32_t ldsOff, const void* gptr,
                                   uint32_t tileW, uint32_t tileH,
                                   uint64_t strideElems, bool padRow) {
  uint64_t ga = (uint64_t)(uintptr_t)gptr;
  u32x4 g0 = {1u,                                   // count=1, load, user
              ldsOff,                               // lds_addr (bytes)
              (uint32_t)ga,                         // global_addr[31:0]
              (uint32_t)((ga >> 32) & 0x1FFFFFFu)   // global_addr[56:32]
                  | (2u << 30)};                    // type=2 ("image")
  uint32_t w0 = (2u << 16);                         // data_size = 4 bytes
  if (padRow) w0 |= (1u << 20)                      // pad_enable
                  | (4u << 22);                     // pad_interval: 32 DWORDs
                                                    // pad_amount=0 -> 1 DWORD
  uint32_t td0 = 1u << 20, td1 = 1u << 20;          // never-clipping dims
  i32x8 g1 = {(int)w0,
              (int)((td0 & 0xFFFFu) << 16),
              (int)(((td0 >> 16) & 0xFFFFu) | ((td1 & 0xFFFFu) << 16)),
              (int)(((td1 >> 16) & 0xFFFFu) | ((tileW & 0xFFFFu) << 16)),
              (int)(tileH & 0xFFFFu),               // tile_dim1; tile_dim2=0
              (int)(uint32_t)strideElems,           // tensor_dim0_stride lo
              (int)((strideElems >> 32) & 0xFFFFu), // stride hi; dim1_stride=0
              0};
  i32x4 z4 = {0, 0, 0, 0};
#if __has_include(<hip/amd_detail/amd_gfx1250_TDM.h>)
  i32x8 z8 = {0, 0, 0, 0, 0, 0, 0, 0};
  __builtin_amdgcn_tensor_load_to_lds(g0, g1, z4, z4, z8, 0);   // clang-23 form
#else
  __builtin_amdgcn_tensor_load_to_lds(g0, g1, z4, z4, 0);       // ROCm 7.2 form
#endif
}

__device__ inline uint32_t lds_off(const void* p) {
  return (uint32_t)(uintptr_t)p;  // low 32 bits of generic ptr = LDS byte offset
}

// ---------------------------------------------------------------------------
// Kernel 1: fused QKV projection.  C[z] = X @ W[z]^T + b[z]
//   grid (M/128, N/128, 3), block 256 (8 waves, 4x2, 32x64 per wave).
//   Both tiles TDM-staged row-major (B[k][n] = W[n][k], so the B fragment is
//   the same adjacent-pair @ stride-33 read as A). Double-buffered.
// ---------------------------------------------------------------------------
__global__ __launch_bounds__(256) void qkv_gemm(
    const float* __restrict__ X,
    const float* __restrict__ Wq, const float* __restrict__ bq,
    const float* __restrict__ Wk, const float* __restrict__ bk,
    const float* __restrict__ Wv, const float* __restrict__ bv,
    float* __restrict__ ws) {
  constexpr int LDT = 33;                 // 32 + TDM pad dword
  __shared__ float As[2][128 * LDT];
  __shared__ float Bs[2][128 * LDT];

  const int z = blockIdx.z;
  const float* W    = (z == 0) ? Wq : (z == 1) ? Wk : Wv;
  const float* bias = (z == 0) ? bq : (z == 1) ? bk : bv;
  float* C = ws + (size_t)z * MN;

  const int row0 = blockIdx.x * 128;
  const int col0 = blockIdx.y * 128;
  const int tid  = threadIdx.x;
  const int wave = tid >> 5, lane = tid & 31;
  const int half = lane >> 4, l16 = lane & 15;
  const int wRow = (wave >> 1) * 32;      // 4 waves along M
  const int wCol = (wave & 1) * 64;       // 2 waves along N

  v8f acc[2][4] = {};

  // prime buffer 0
  if (tid == 0) {
    tdm_load_2d(lds_off(&As[0][0]), &X[(size_t)row0 * DMODEL], 32, 128, DMODEL, true);
    tdm_load_2d(lds_off(&Bs[0][0]), &W[(size_t)col0 * DMODEL], 32, 128, DMODEL, true);
  }

  int buf = 0;
  for (int k0 = 0; k0 < DMODEL; k0 += 32) {
    const bool more = (k0 + 32) < DMODEL;
    if (tid == 0) {
      if (more) {   // kick DMA for next tile pair into the other buffer
        tdm_load_2d(lds_off(&As[buf ^ 1][0]),
                    &X[(size_t)row0 * DMODEL + k0 + 32], 32, 128, DMODEL, true);
        tdm_load_2d(lds_off(&Bs[buf ^ 1][0]),
                    &W[(size_t)col0 * DMODEL + k0 + 32], 32, 128, DMODEL, true);
        __builtin_amdgcn_s_wait_tensorcnt(2);  // in-order: current pair landed
      } else {
        __builtin_amdgcn_s_wait_tensorcnt(0);
      }
    }
    __syncthreads();

    const float* Acur = &As[buf][0];
    const float* Bcur = &Bs[buf][0];
#pragma unroll
    for (int kk = 0; kk < 32; kk += 4) {
      v2f a[2], b[4];
#pragma unroll
      for (int mi = 0; mi < 2; ++mi) {
        const float* p = &Acur[(wRow + mi * 16 + l16) * LDT + kk + 2 * half];
        a[mi].x = p[0]; a[mi].y = p[1];          // A[m][k], A[m][k+1]
      }
#pragma unroll
      for (int ni = 0; ni < 4; ++ni) {
        const float* p = &Bcur[(wCol + ni * 16 + l16) * LDT + kk + 2 * half];
        b[ni].x = p[0]; b[ni].y = p[1];          // B[k][n]=W[n][k], B[k+1][n]
      }
#pragma unroll
      for (int mi = 0; mi < 2; ++mi)
#pragma unroll
        for (int ni = 0; ni < 4; ++ni)
          acc[mi][ni] = WMMA_F32(a[mi], b[ni], acc[mi][ni]);
    }
    __syncthreads();   // all waves done with buf before TDM overwrites it
    buf ^= 1;
  }

  // Epilogue: + bias.  C frag element r -> (M = 8*half + r, N = l16)
#pragma unroll
  for (int ni = 0; ni < 4; ++ni) {
    int n = col0 + wCol + ni * 16 + l16;
    float bv_ = bias[n];
#pragma unroll
    for (int mi = 0; mi < 2; ++mi)
#pragma unroll
      for (int r = 0; r < 8; ++r) {
        int m = row0 + wRow + mi * 16 + 8 * half + r;
        C[(size_t)m * DMODEL + n] = acc[mi][ni][r] + bv_;
      }
  }
}

// ---------------------------------------------------------------------------
// Kernel 2: Linformer projection.  Out[bh] = P @ Head(Src)[bh]
//   P (E or D): [128 x 4096], Src: [B*S x 1024] head-interleaved.
//   grid (64), block 256 (8 waves 4x2, 32x32 per wave). TDM double-buffered.
// ---------------------------------------------------------------------------
__global__ __launch_bounds__(256) void proj_gemm(
    const float* __restrict__ P, const float* __restrict__ Src,
    float* __restrict__ Out) {
  constexpr int LDA = 33;
  __shared__ float Es[2][128 * LDA];
  __shared__ float Bs[2][32 * 64];

  const int bh = blockIdx.x;
  const int b = bh >> 4, h = bh & 15;
  const int tid = threadIdx.x;
  const int wave = tid >> 5, lane = tid & 31;
  const int half = lane >> 4, l16 = lane & 15;
  const int wRow = (wave >> 1) * 32;
  const int wCol = (wave & 1) * 32;
  float* O = Out + (size_t)bh * KDIM * DHEAD;

  v8f acc[2][2] = {};

  if (tid == 0) {
    tdm_load_2d(lds_off(&Es[0][0]), &P[0], 32, 128, SEQ, true);
    tdm_load_2d(lds_off(&Bs[0][0]),
                &Src[((size_t)b * SEQ) * DMODEL + h * DHEAD], 64, 32, DMODEL, false);
  }

  int buf = 0;
  for (int s0 = 0; s0 < SEQ; s0 += 32) {
    const bool more = (s0 + 32) < SEQ;
    if (tid == 0) {
      if (more) {
        tdm_load_2d(lds_off(&Es[buf ^ 1][0]), &P[s0 + 32], 32, 128, SEQ, true);
        tdm_load_2d(lds_off(&Bs[buf ^ 1][0]),
                    &Src[((size_t)b * SEQ + s0 + 32) * DMODEL + h * DHEAD],
                    64, 32, DMODEL, false);
        __builtin_amdgcn_s_wait_tensorcnt(2);
      } else {
        __builtin_amdgcn_s_wait_tensorcnt(0);
      }
    }
    __syncthreads();

    const float* Ecur = &Es[buf][0];
    const float* Bcur = &Bs[buf][0];
#pragma unroll
    for (int kk = 0; kk < 32; kk += 4) {
      v2f a[2], bb[2];
#pragma unroll
      for (int mi = 0; mi < 2; ++mi) {
        const float* p = &Ecur[(wRow + mi * 16 + l16) * LDA + kk + 2 * half];
        a[mi].x = p[0]; a[mi].y = p[1];
      }
#pragma unroll
      for (int ni = 0; ni < 2; ++ni) {
        const float* p = &Bcur[(kk + 2 * half) * 64 + wCol + ni * 16 + l16];
        bb[ni].x = p[0]; bb[ni].y = p[64];
      }
#pragma unroll
      for (int mi = 0; mi < 2; ++mi)
#pragma unroll
        for (int ni = 0; ni < 2; ++ni)
          acc[mi][ni] = WMMA_F32(a[mi], bb[ni], acc[mi][ni]);
    }
    __syncthreads();
    buf ^= 1;
  }

#pragma unroll
  for (int mi = 0; mi < 2; ++mi)
#pragma unroll
    for (int ni = 0; ni < 2; ++ni)
#pragma unroll
      for (int r = 0; r < 8; ++r)
        O[(size_t)(wRow + mi * 16 + 8 * half + r) * DHEAD +
          wCol + ni * 16 + l16] = acc[mi][ni][r];
}

// ---------------------------------------------------------------------------
// Kernel 3: attention core per (b,h, 128-row S block):
//   scores = Q*pk^T / 8 (WMMA) -> softmax over 128 -> ctx = probs*pv (WMMA)
//   grid (S/128=32, B*H=64), block 256
// ---------------------------------------------------------------------------
__global__ __launch_bounds__(256) void attn_core(
    const float* __restrict__ Q, const float* __restrict__ pk,
    const float* __restrict__ pv, float* __restrict__ out) {
  constexpr int LDS_SC = 129, LDQ = 68, LDPK = 129;
  __shared__ float smem[128 * LDS_SC + 128 * LDQ + 64 * LDPK];  // 133,888 B
  float* sc  = smem;                       // 128 x 129 scores/probs
  float* Qs  = smem + 128 * LDS_SC;        // 128 x 68  (phase 1)
  float* pkT = Qs + 128 * LDQ;             // 64  x 129 (phase 1)
  float* pvS = smem + 128 * LDS_SC;        // 128 x 64  (phase 3, reuses Qs)

  const int s0 = blockIdx.x * 128;
  const int bh = blockIdx.y;
  const int b = bh >> 4, h = bh & 15;
  const int tid = threadIdx.x;
  const int wave = tid >> 5, lane = tid & 31;
  const int half = lane >> 4, l16 = lane & 15;
  const int wRow = (wave >> 1) * 32;
  const float* PK = pk + (size_t)bh * KDIM * DHEAD;
  const float* PV = pv + (size_t)bh * KDIM * DHEAD;

  // ---- stage Q block (128x64) and pk^T (64x128) ----
  for (int i = tid; i < 2048; i += 256) {
    int r = i >> 4, c4 = (i & 15) * 4;
    float4 vv = *(const float4*)
        &Q[((size_t)b * SEQ + s0 + r) * DMODEL + h * DHEAD + c4];
    *(float4*)&Qs[r * LDQ + c4] = vv;
  }
  for (int i = tid; i < 2048; i += 256) {
    int kk = i >> 4, c4 = (i & 15) * 4;
    float4 vv = *(const float4*)&PK[kk * DHEAD + c4];
    pkT[(c4 + 0) * LDPK + kk] = vv.x;
    pkT[(c4 + 1) * LDPK + kk] = vv.y;
    pkT[(c4 + 2) * LDPK + kk] = vv.z;
    pkT[(c4 + 3) * LDPK + kk] = vv.w;
  }
  __syncthreads();

  // ---- phase 1: scores (M=128, N=128, K=64), wave tile 32x64 ----
  {
    const int wCol = (wave & 1) * 64;
    v8f acc[2][4] = {};
#pragma unroll
    for (int kk = 0; kk < DHEAD; kk += 4) {
      v2f a[2], bb[4];
#pragma unroll
      for (int mi = 0; mi < 2; ++mi) {
        const float* p = &Qs[(wRow + mi * 16 + l16) * LDQ + kk + 2 * half];
        a[mi].x = p[0]; a[mi].y = p[1];
      }
#pragma unroll
      for (int ni = 0; ni < 4; ++ni) {
        const float* p = &pkT[(kk + 2 * half) * LDPK + wCol + ni * 16 + l16];
        bb[ni].x = p[0]; bb[ni].y = p[LDPK];
      }
#pragma unroll
      for (int mi = 0; mi < 2; ++mi)
#pragma unroll
        for (int ni = 0; ni < 4; ++ni)
          acc[mi][ni] = WMMA_F32(a[mi], bb[ni], acc[mi][ni]);
    }
#pragma unroll
    for (int mi = 0; mi < 2; ++mi)
#pragma unroll
      for (int ni = 0; ni < 4; ++ni)
#pragma unroll
        for (int r = 0; r < 8; ++r)
          sc[(wRow + mi * 16 + 8 * half + r) * LDS_SC + wCol + ni * 16 + l16] =
              acc[mi][ni][r] * 0.125f;               // 1/sqrt(64)
  }
  __syncthreads();

  // ---- stage pv (overwrites Qs region), then softmax in sc ----
  for (int i = tid; i < 2048; i += 256) {
    int kk = i >> 4, c4 = (i & 15) * 4;
    *(float4*)&pvS[kk * 64 + c4] = *(const float4*)&PV[kk * DHEAD + c4];
  }
  if (tid < 128) {
    float* row = sc + tid * LDS_SC;
    float mx = row[0];
    for (int j = 1; j < KDIM; ++j) mx = fmaxf(mx, row[j]);
    float sum = 0.f;
    for (int j = 0; j < KDIM; ++j) { float e = __expf(row[j] - mx); row[j] = e; sum += e; }
    float inv = 1.0f / sum;
    for (int j = 0; j < KDIM; ++j) row[j] *= inv;
  }
  __syncthreads();

  // ---- phase 3: ctx = probs @ pv (M=128, N=64, K=128), wave tile 32x32 ----
  {
    const int wCol = (wave & 1) * 32;
    v8f acc[2][2] = {};
#pragma unroll 8
    for (int kk = 0; kk < KDIM; kk += 4) {
      v2f a[2], bb[2];
#pragma unroll
      for (int mi = 0; mi < 2; ++mi) {
        const float* p = &sc[(wRow + mi * 16 + l16) * LDS_SC + kk + 2 * half];
        a[mi].x = p[0]; a[mi].y = p[1];
      }
#pragma unroll
      for (int ni = 0; ni < 2; ++ni) {
        const float* p = &pvS[(kk + 2 * half) * 64 + wCol + ni * 16 + l16];
        bb[ni].x = p[0]; bb[ni].y = p[64];
      }
#pragma unroll
      for (int mi = 0; mi < 2; ++mi)
#pragma unroll
        for (int ni = 0; ni < 2; ++ni)
          acc[mi][ni] = WMMA_F32(a[mi], bb[ni], acc[mi][ni]);
    }
#pragma unroll
    for (int mi = 0; mi < 2; ++mi)
#pragma unroll
      for (int ni = 0; ni < 2; ++ni)
#pragma unroll
        for (int r = 0; r < 8; ++r) {
          int m = wRow + mi * 16 + 8 * half + r;
          int n = wCol + ni * 16 + l16;
          out[((size_t)b * SEQ + s0 + m) * DMODEL + h * DHEAD + n] =
              acc[mi][ni][r];
        }
  }
}

// ---------------------------------------------------------------------------
extern "C" void kernel_launch(void* const* d_in, const int* in_sizes, int n_in,
                              void* d_out, int out_size, void* d_ws, size_t ws_size,
                              hipStream_t stream) {
  const float* X  = (const float*)d_in[0];
  const float* Wq = (const float*)d_in[1];
  const float* bq = (const float*)d_in[2];
  const float* Wk = (const float*)d_in[3];
  const float* bk = (const float*)d_in[4];
  const float* Wv = (const float*)d_in[5];
  const float* bv = (const float*)d_in[6];
  const float* E  = (const float*)d_in[7];
  const float* D  = (const float*)d_in[8];

  float* ws = (float*)d_ws;
  float* q  = ws;                 // [B*S, DM]
  float* k  = ws + MN;
  float* v  = ws + 2 * MN;
  float* pk = ws + 3 * MN;        // [B*H, 128, 64]
  float* pv = pk + (size_t)BATCH * NH * KDIM * DHEAD;

  qkv_gemm<<<dim3(BATCH * SEQ / 128, DMODEL / 128, 3), 256, 0, stream>>>(
      X, Wq, bq, Wk, bk, Wv, bv, ws);
  proj_gemm<<<dim3(BATCH * NH), 256, 0, stream>>>(E, k, pk);
  proj_gemm<<<dim3(BATCH * NH), 256, 0, stream>>>(D, v, pv);
  attn_core<<<dim3(SEQ / 128, BATCH * NH), 256, 0, stream>>>(
      q, pk, pv, (float*)d_out);
}